// AttentiveGRU1_5136780886781
// MI455X (gfx1250) — compile-verified
//
#include <hip/hip_runtime.h>
#include <hip/hip_bf16.h>
#include <math.h>

typedef __attribute__((ext_vector_type(2))) float v2f;
typedef __attribute__((ext_vector_type(8))) float v8f;

#define EDGE_FEAT 32
#define EDGE_HIDDEN 64
#define NODE_FEAT 128

static __device__ __forceinline__ v8f wmma4(v2f a, v2f b, v8f c) {
    // D = A(16x4 f32) x B(4x16 f32) + C(16x16 f32)
    return __builtin_amdgcn_wmma_f32_16x16x4_f32(
        false, a, false, b, (short)0, c, false, false);
}

// order-preserving uint encoding of float for atomic max
static __device__ __forceinline__ unsigned fenc(float f) {
    unsigned u = __float_as_uint(f);
    return (u & 0x80000000u) ? ~u : (u | 0x80000000u);
}
static __device__ __forceinline__ float fdec(unsigned e) {
    return __uint_as_float((e & 0x80000000u) ? (e & 0x7FFFFFFFu) : ~e);
}

static __device__ __forceinline__ float sigmoidf_(float x) {
    return 1.0f / (1.0f + __expf(-x));
}

// ---------------- kernel 1: init scratch -------------------------------------
__global__ __launch_bounds__(256) void k_init(float* __restrict__ c,
                                              float* __restrict__ seg_sum,
                                              unsigned* __restrict__ seg_max,
                                              int N) {
    int i = blockIdx.x * blockDim.x + threadIdx.x;
    int total = N * EDGE_HIDDEN;
    if (i < total) c[i] = 0.0f;
    if (i < N) { seg_sum[i] = 0.0f; seg_max[i] = 0u; }
}

// ---------------- kernel 2: segment max --------------------------------------
__global__ __launch_bounds__(256) void k_segmax(const float* __restrict__ logits,
                                                const int* __restrict__ dst,
                                                unsigned* __restrict__ seg_max,
                                                int E) {
    int e = blockIdx.x * blockDim.x + threadIdx.x;
    if (e >= E) return;
    atomicMax(&seg_max[dst[e]], fenc(logits[e]));
}

// ---------------- kernel 3: exp + segment sum --------------------------------
__global__ __launch_bounds__(256) void k_expsum(const float* __restrict__ logits,
                                                const int* __restrict__ dst,
                                                const unsigned* __restrict__ seg_max,
                                                float* __restrict__ ex,
                                                float* __restrict__ seg_sum,
                                                int E) {
    int e = blockIdx.x * blockDim.x + threadIdx.x;
    if (e >= E) return;
    int d = dst[e];
    float v = __expf(logits[e] - fdec(seg_max[d]));
    ex[e] = v;
    atomicAdd(&seg_sum[d], v);
}

// ---------------- kernel 4: edge GEMM (WMMA) + attention scatter -------------
// 16 edges per wave, 8 waves per block -> 128 edges/block. E % 128 == 0.
__global__ __launch_bounds__(256) void k_edge_gemm(
    const float* __restrict__ edge_feats,   // [E,32]
    const float* __restrict__ W_e,          // [64,32]
    const float* __restrict__ b_e,          // [64]
    const float* __restrict__ ex,           // [E]
    const float* __restrict__ seg_sum,      // [N]
    const int* __restrict__ dst,            // [E]
    float* __restrict__ c)                  // [N,64]
{
    const int lane = threadIdx.x & 31;
    const int wv   = threadIdx.x >> 5;
    const int lo   = lane & 15;
    const int hi   = lane >> 4;
    const long m0  = ((long)blockIdx.x * 8 + wv) * 16;

    const float* arow = edge_feats + (m0 + lo) * EDGE_FEAT + 2 * hi;

    v8f acc0 = {}, acc1 = {}, acc2 = {}, acc3 = {};
#pragma unroll
    for (int kk = 0; kk < 8; ++kk) {
        v2f a = *(const v2f*)(arow + kk * 4);
        const float* wb = W_e + kk * 4 + 2 * hi;
        v2f b0 = *(const v2f*)(wb + (lo +  0) * EDGE_FEAT);
        v2f b1 = *(const v2f*)(wb + (lo + 16) * EDGE_FEAT);
        v2f b2 = *(const v2f*)(wb + (lo + 32) * EDGE_FEAT);
        v2f b3 = *(const v2f*)(wb + (lo + 48) * EDGE_FEAT);
        acc0 = wmma4(a, b0, acc0);
        acc1 = wmma4(a, b1, acc1);
        acc2 = wmma4(a, b2, acc2);
        acc3 = wmma4(a, b3, acc3);
    }

    float be0 = b_e[lo], be1 = b_e[lo + 16], be2 = b_e[lo + 32], be3 = b_e[lo + 48];
#pragma unroll
    for (int i = 0; i < 8; ++i) {
        long m = m0 + i + 8 * hi;            // edge index for this D element
        int d = dst[m];
        float alpha = ex[m] / seg_sum[d];
        float* crow = c + (long)d * EDGE_HIDDEN;
        atomicAdd(crow + lo +  0, alpha * (acc0[i] + be0));
        atomicAdd(crow + lo + 16, alpha * (acc1[i] + be1));
        atomicAdd(crow + lo + 32, alpha * (acc2[i] + be2));
        atomicAdd(crow + lo + 48, alpha * (acc3[i] + be3));
    }
}

// ---------------- kernel 5: fused GRU node update (WMMA) ---------------------
// 16 nodes per wave, 5 waves per block. N == 3125*16, 3125 == 625*5 (no tail).
#define CTX_LD 68    // 64 padded; 68 % 64 == 4 -> conflict-free dual-bank reads
#define H_LD   132   // 128 padded; 132 % 64 == 4
__global__ __launch_bounds__(160) void k_node_gru(
    const float* __restrict__ cbuf,         // [N,64]  (pre-elu context accum)
    const float* __restrict__ node_feats,   // [N,128]
    const float* __restrict__ W_ih,         // [384,64]
    const float* __restrict__ W_hh,         // [384,128]
    const float* __restrict__ b_ih,         // [384]
    const float* __restrict__ b_hh,         // [384]
    float* __restrict__ out)                // [N,128]
{
    __shared__ float s_ctx[5 * 16 * CTX_LD];
    __shared__ float s_h  [5 * 16 * H_LD];

    const int lane = threadIdx.x & 31;
    const int wv   = threadIdx.x >> 5;
    const int lo   = lane & 15;
    const int hi   = lane >> 4;
    const long m0  = ((long)blockIdx.x * 5 + wv) * 16;

    float* sc = s_ctx + wv * 16 * CTX_LD;
    float* sh = s_h   + wv * 16 * H_LD;

    // stage elu(context) rows
    for (int idx = lane; idx < 16 * EDGE_HIDDEN; idx += 32) {
        int r = idx >> 6, col = idx & 63;
        float x = cbuf[(m0 + r) * EDGE_HIDDEN + col];
        sc[r * CTX_LD + col] = x > 0.0f ? x : (__expf(x) - 1.0f);   // elu
    }
    // stage h rows
    for (int idx = lane; idx < 16 * NODE_FEAT; idx += 32) {
        int r = idx >> 7, col = idx & 127;
        sh[r * H_LD + col] = node_feats[(m0 + r) * NODE_FEAT + col];
    }
    __syncthreads();

    const float* scA = sc + lo * CTX_LD + 2 * hi;
    const float* shA = sh + lo * H_LD   + 2 * hi;

#pragma unroll 1
    for (int jt = 0; jt < 8; ++jt) {
        const int jr = jt * 16 + lo;          // r-gate column (0..127)
        const int jz = 128 + jr;              // z-gate column
        const int jn = 256 + jr;              // n-gate column

        v8f accr = {}, accz = {}, gin = {}, ghn = {};

        // context sweep (K = 64): contributes to r, z, i_n
#pragma unroll
        for (int kk = 0; kk < 16; ++kk) {
            v2f a = *(const v2f*)(scA + kk * 4);
            const float* wb = W_ih + kk * 4 + 2 * hi;
            accr = wmma4(a, *(const v2f*)(wb + (long)jr * EDGE_HIDDEN), accr);
            accz = wmma4(a, *(const v2f*)(wb + (long)jz * EDGE_HIDDEN), accz);
            gin  = wmma4(a, *(const v2f*)(wb + (long)jn * EDGE_HIDDEN), gin);
        }
        // hidden sweep (K = 128): contributes to r, z, h_n
#pragma unroll
        for (int kk = 0; kk < 32; ++kk) {
            v2f a = *(const v2f*)(shA + kk * 4);
            const float* wb = W_hh + kk * 4 + 2 * hi;
            accr = wmma4(a, *(const v2f*)(wb + (long)jr * NODE_FEAT), accr);
            accz = wmma4(a, *(const v2f*)(wb + (long)jz * NODE_FEAT), accz);
            ghn  = wmma4(a, *(const v2f*)(wb + (long)jn * NODE_FEAT), ghn);
        }

        const float brz = b_ih[jr] + b_hh[jr];
        const float bzz = b_ih[jz] + b_hh[jz];
        const float bin = b_ih[jn];
        const float bhn = b_hh[jn];

#pragma unroll
        for (int i = 0; i < 8; ++i) {
            int row = i + 8 * hi;                 // local node row (0..15)
            float r = sigmoidf_(accr[i] + brz);
            float z = sigmoidf_(accz[i] + bzz);
            float n = tanhf(gin[i] + bin + r * (ghn[i] + bhn));
            float h = sh[row * H_LD + jt * 16 + lo];
            float hn = (1.0f - z) * n + z * h;
            out[(m0 + row) * NODE_FEAT + jt * 16 + lo] = hn > 0.0f ? hn : 0.0f;
        }
    }
}

// ---------------- launcher ---------------------------------------------------
extern "C" void kernel_launch(void* const* d_in, const int* in_sizes, int n_in,
                              void* d_out, int out_size, void* d_ws, size_t ws_size,
                              hipStream_t stream) {
    const float* edge_logits = (const float*)d_in[0];   // [E,1]
    const float* edge_feats  = (const float*)d_in[1];   // [E,32]
    const float* node_feats  = (const float*)d_in[2];   // [N,128]
    const int*   dst         = (const int*)  d_in[3];   // [E]
    const float* W_e         = (const float*)d_in[4];   // [64,32]
    const float* b_e         = (const float*)d_in[5];   // [64]
    const float* W_ih        = (const float*)d_in[6];   // [384,64]
    const float* W_hh        = (const float*)d_in[7];   // [384,128]
    const float* b_ih        = (const float*)d_in[8];   // [384]
    const float* b_hh        = (const float*)d_in[9];   // [384]
    float* out = (float*)d_out;

    const int E = in_sizes[0];
    const int N = in_sizes[2] / NODE_FEAT;

    // workspace layout
    char* ws = (char*)d_ws;
    float*    seg_sum = (float*)ws;                ws += (size_t)N * sizeof(float);
    unsigned* seg_max = (unsigned*)ws;             ws += (size_t)N * sizeof(unsigned);
    float*    ex      = (float*)ws;                ws += (size_t)E * sizeof(float);
    float*    c       = (float*)ws;                // [N,64]

    // 1) init
    {
        int total = N * EDGE_HIDDEN;
        int nb = (total + 255) / 256;
        k_init<<<nb, 256, 0, stream>>>(c, seg_sum, seg_max, N);
    }
    // 2) segment max
    k_segmax<<<(E + 255) / 256, 256, 0, stream>>>(edge_logits, dst, seg_max, E);
    // 3) exp + segment sum
    k_expsum<<<(E + 255) / 256, 256, 0, stream>>>(edge_logits, dst, seg_max, ex, seg_sum, E);
    // 4) edge GEMM + attention-weighted scatter (E divisible by 128)
    k_edge_gemm<<<E / 128, 256, 0, stream>>>(edge_feats, W_e, b_e, ex, seg_sum, dst, c);
    // 5) fused GRU node update (N divisible by 80 = 5 waves * 16 rows)
    k_node_gru<<<N / 80, 160, 0, stream>>>(c, node_feats, W_ih, W_hh, b_ih, b_hh, out);
}